// FlexGCN_15659450761918
// MI455X (gfx1250) — compile-verified
//
#include <hip/hip_runtime.h>
#include <math.h>

// ---------------------------------------------------------------------------
// FlexGCN on gfx1250 (MI455X): GIN convs via V_WMMA_F32_16X16X4_F32,
// edge aggregation via global f32 atomics, per-graph bitonic top-k in LDS.
// ---------------------------------------------------------------------------

typedef float v2f __attribute__((ext_vector_type(2)));
typedef float v8f __attribute__((ext_vector_type(8)));

#define G_GRAPHS   64
#define N_NODES    1024
#define E_TOTAL    1048576   // 64 * 16384
#define DIN        128

// ---------------------------------------------------------------------------
// Scatter aggregation: AGG[dst[e]] += X[src[e]] (optionally masked).
// One thread handles one (edge, 4-feature-chunk); f32 global atomics.
// ---------------------------------------------------------------------------
template <int C>
__global__ __launch_bounds__(256) void scatter_agg_kernel(
    const float* __restrict__ X, const int* __restrict__ src,
    const int* __restrict__ dst, const int* __restrict__ msk,
    float* __restrict__ AGG, int E) {
  constexpr int CH = C / 4;
  long long idx = (long long)blockIdx.x * blockDim.x + threadIdx.x;
  long long total = (long long)E * CH;
  if (idx >= total) return;
  int e = (int)(idx / CH);
  int f = (int)(idx % CH) * 4;
  if (msk && msk[e] == 0) return;
  int s = src[e];
  int d = dst[e];
  const float4 v = *(const float4*)(X + (size_t)s * C + f);
  float* o = AGG + (size_t)d * C + f;
  atomicAdd(o + 0, v.x);
  atomicAdd(o + 1, v.y);
  atomicAdd(o + 2, v.z);
  atomicAdd(o + 3, v.w);
}

// ---------------------------------------------------------------------------
// Fused GIN linear: OUT = relu((X + AGG) @ W + bias), via f32 WMMA.
// One wave owns a 16-row x N output strip; K consumed 4 at a time.
//
// V_WMMA_F32_16X16X4_F32 fragment layout (wave32):
//   A (16x4): lanes 0-15 -> M=lane, VGPR0=K0, VGPR1=K1; lanes 16-31 -> K2,K3.
//   B (4x16): lanes 0-15 -> N=lane, VGPR0=K0, VGPR1=K1; lanes 16-31 -> K2,K3.
//   C/D (16x16): VGPR r, lane l -> M = r + 8*(l>=16), N = l&15.
// ---------------------------------------------------------------------------
template <int K, int N>
__global__ __launch_bounds__(128) void gin_gemm_relu_kernel(
    const float* __restrict__ X, const float* __restrict__ AGG,
    const float* __restrict__ W, const float* __restrict__ bias,
    float* __restrict__ OUT) {
  constexpr int NT = N / 16;
  const int lane = threadIdx.x & 31;
  const int wave = threadIdx.x >> 5;
  const int l16  = lane & 15;
  const int half = lane >> 4;           // 0: K pair {0,1}, 1: K pair {2,3}
  const int m0   = (blockIdx.x * 4 + wave) * 16;

  v8f acc[NT] = {};

  const int arow = m0 + l16;
  const float* __restrict__ xr = X   + (size_t)arow * K;
  const float* __restrict__ ar = AGG + (size_t)arow * K;

  for (int k = 0; k < K; k += 4) {
    const int kb = k + 2 * half;
    v2f a;
    a.x = xr[kb]     + ar[kb];
    a.y = xr[kb + 1] + ar[kb + 1];
#pragma unroll
    for (int nt = 0; nt < NT; ++nt) {
      const int col = nt * 16 + l16;
      v2f b;
      b.x = W[(size_t)kb * N + col];
      b.y = W[(size_t)(kb + 1) * N + col];
      acc[nt] = __builtin_amdgcn_wmma_f32_16x16x4_f32(
          /*neg_a=*/false, a, /*neg_b=*/false, b,
          /*c_mod=*/(short)0, acc[nt], /*reuse_a=*/false, /*reuse_b=*/false);
    }
  }

#pragma unroll
  for (int nt = 0; nt < NT; ++nt) {
    const int n  = nt * 16 + l16;
    const float bv = bias[n];
#pragma unroll
    for (int r = 0; r < 8; ++r) {
      const int m = m0 + half * 8 + r;
      float v = acc[nt][r] + bv;
      OUT[(size_t)m * N + n] = v > 0.0f ? v : 0.0f;
    }
  }
}

// ---------------------------------------------------------------------------
// TopKPooling(ratio=0.5): per-graph score = tanh(x.p/||p||), bitonic sort of
// (score, idx) pairs in LDS (descending, index tie-break), keep top n/2,
// gate kept features by score, record old->new remap (if requested).
// blockDim.x must equal n (power of two <= 1024); one block per graph.
// ---------------------------------------------------------------------------
template <int C>
__global__ __launch_bounds__(1024) void topk_pool_kernel(
    const float* __restrict__ X, const float* __restrict__ p,
    float* __restrict__ XOUT, int* __restrict__ remap, int n) {
  __shared__ float s_val[1024];
  __shared__ int   s_idx[1024];

  const int g   = blockIdx.x;
  const int tid = threadIdx.x;

  // score for node (g, tid)
  float pn = 0.0f;
#pragma unroll
  for (int c = 0; c < C; ++c) pn += p[c] * p[c];
  pn = sqrtf(pn);
  const float* xrow = X + (size_t)(g * n + tid) * C;
  float dot = 0.0f;
#pragma unroll
  for (int c = 0; c < C; ++c) dot += xrow[c] * p[c];
  s_val[tid] = tanhf(dot / pn);
  s_idx[tid] = tid;
  __syncthreads();

  // bitonic sort, descending by value, ascending index on ties
  for (int kk = 2; kk <= n; kk <<= 1) {
    for (int j = kk >> 1; j > 0; j >>= 1) {
      const int ixj = tid ^ j;
      if (ixj > tid) {
        const float v1 = s_val[tid], v2 = s_val[ixj];
        const int   i1 = s_idx[tid], i2 = s_idx[ixj];
        const bool firstBetter = (v1 > v2) || (v1 == v2 && i1 < i2);
        const bool up = ((tid & kk) == 0);        // "ascending" = best first
        if (up ? !firstBetter : firstBetter) {
          s_val[tid] = v2; s_val[ixj] = v1;
          s_idx[tid] = i2; s_idx[ixj] = i1;
        }
      }
      __syncthreads();
    }
  }

  const int k = n >> 1;
  if (remap && tid < k) remap[g * n + s_idx[tid]] = g * k + tid;

  // gated feature copy for kept nodes
  for (int t = tid; t < k * C; t += blockDim.x) {
    const int j = t / C, f = t % C;
    XOUT[(size_t)(g * k + j) * C + f] =
        X[(size_t)(g * n + s_idx[j]) * C + f] * s_val[j];
  }
}

// ---------------------------------------------------------------------------
// Edge remap/filter after pooling (pool-1 only; original mask is all-true).
// ---------------------------------------------------------------------------
__global__ __launch_bounds__(256) void remap_edges_kernel(
    const int* __restrict__ src, const int* __restrict__ dst,
    const int* __restrict__ remap, int* __restrict__ s2,
    int* __restrict__ d2, int* __restrict__ m2, int E) {
  int e = blockIdx.x * blockDim.x + threadIdx.x;
  if (e >= E) return;
  const int s = remap[src[e]];
  const int d = remap[dst[e]];
  const int ok = (s >= 0) && (d >= 0);
  s2[e] = ok ? s : 0;
  d2[e] = ok ? d : 0;
  m2[e] = ok;
}

// ---------------------------------------------------------------------------
// Global mean pool (64 graphs x 256 nodes x 16 feats) + 16->2 classifier.
// ---------------------------------------------------------------------------
__global__ __launch_bounds__(32) void mean_classify_kernel(
    const float* __restrict__ X3, const float* __restrict__ Wc,
    const float* __restrict__ bc, float* __restrict__ out) {
  __shared__ float pooled[16];
  const int g = blockIdx.x, t = threadIdx.x;
  if (t < 16) {
    float s = 0.0f;
    for (int i = 0; i < 256; ++i) s += X3[(size_t)(g * 256 + i) * 16 + t];
    pooled[t] = s * (1.0f / 256.0f);
  }
  __syncthreads();
  if (t < 2) {
    float s = bc[t];
#pragma unroll
    for (int f = 0; f < 16; ++f) s += pooled[f] * Wc[f * 2 + t];
    out[g * 2 + t] = s;
  }
}

// ---------------------------------------------------------------------------
// Launch
// ---------------------------------------------------------------------------
static inline int cdiv(long long a, long long b) { return (int)((a + b - 1) / b); }

extern "C" void kernel_launch(void* const* d_in, const int* in_sizes, int n_in,
                              void* d_out, int out_size, void* d_ws, size_t ws_size,
                              hipStream_t stream) {
  const float* x   = (const float*)d_in[0];
  const int*   ei  = (const int*)d_in[1];
  const float* W0  = (const float*)d_in[2];
  const float* b0  = (const float*)d_in[3];
  const float* W1  = (const float*)d_in[4];
  const float* b1  = (const float*)d_in[5];
  const float* W2  = (const float*)d_in[6];
  const float* b2  = (const float*)d_in[7];
  const float* p1  = (const float*)d_in[8];
  const float* p2  = (const float*)d_in[9];
  const float* Wc  = (const float*)d_in[10];
  const float* bc  = (const float*)d_in[11];

  const int E = in_sizes[1] / 2;               // 1048576
  const int* src = ei;
  const int* dst = ei + E;

  const int M0 = G_GRAPHS * N_NODES;           // 65536 nodes
  const int M2 = G_GRAPHS * (N_NODES / 2);     // 32768 nodes after pool1

  // workspace carve-up (~95 MB)
  char* w = (char*)d_ws;
  float* agg0 = (float*)w; w += (size_t)M0 * 128 * 4;
  float* h0   = (float*)w; w += (size_t)M0 * 64 * 4;
  float* agg1 = (float*)w; w += (size_t)M0 * 64 * 4;
  float* h1   = (float*)w; w += (size_t)M0 * 32 * 4;
  float* x2   = (float*)w; w += (size_t)M2 * 32 * 4;
  int*   rmp  = (int*)w;   w += (size_t)M0 * 4;
  int*   s2   = (int*)w;   w += (size_t)E * 4;
  int*   d2   = (int*)w;   w += (size_t)E * 4;
  int*   m2   = (int*)w;   w += (size_t)E * 4;
  float* agg2 = (float*)w; w += (size_t)M2 * 32 * 4;
  float* h2   = (float*)w; w += (size_t)M2 * 16 * 4;
  float* x3   = (float*)w; w += (size_t)(M2 / 2) * 16 * 4;

  hipMemsetAsync(agg0, 0,    (size_t)M0 * 128 * 4, stream);
  hipMemsetAsync(agg1, 0,    (size_t)M0 * 64 * 4,  stream);
  hipMemsetAsync(agg2, 0,    (size_t)M2 * 32 * 4,  stream);
  hipMemsetAsync(rmp,  0xFF, (size_t)M0 * 4,       stream);   // -1

  // stem: GIN 128->64 + ReLU
  scatter_agg_kernel<128><<<cdiv((long long)E * 32, 256), 256, 0, stream>>>(
      x, src, dst, nullptr, agg0, E);
  gin_gemm_relu_kernel<128, 64><<<M0 / 64, 128, 0, stream>>>(x, agg0, W0, b0, h0);

  // layer 1: GIN 64->32 + ReLU
  scatter_agg_kernel<64><<<cdiv((long long)E * 16, 256), 256, 0, stream>>>(
      h0, src, dst, nullptr, agg1, E);
  gin_gemm_relu_kernel<64, 32><<<M0 / 64, 128, 0, stream>>>(h0, agg1, W1, b1, h1);

  // pool 1: keep 512/1024 per graph, remap edges
  topk_pool_kernel<32><<<G_GRAPHS, N_NODES, 0, stream>>>(h1, p1, x2, rmp, N_NODES);
  remap_edges_kernel<<<cdiv(E, 256), 256, 0, stream>>>(src, dst, rmp, s2, d2, m2, E);

  // layer 2: GIN 32->16 + ReLU (masked edges)
  scatter_agg_kernel<32><<<cdiv((long long)E * 8, 256), 256, 0, stream>>>(
      x2, s2, d2, m2, agg2, E);
  gin_gemm_relu_kernel<32, 16><<<M2 / 64, 128, 0, stream>>>(x2, agg2, W2, b2, h2);

  // pool 2: keep 256/512 per graph (edges unused afterwards)
  topk_pool_kernel<16><<<G_GRAPHS, N_NODES / 2, 0, stream>>>(h2, p2, x3, nullptr,
                                                             N_NODES / 2);

  // mean pool + classifier -> [64, 2]
  mean_classify_kernel<<<G_GRAPHS, 32, 0, stream>>>(x3, Wc, bc, (float*)d_out);
}